// LSTMScalar_82145544503297
// MI455X (gfx1250) — compile-verified
//
#include <hip/hip_runtime.h>

typedef __attribute__((ext_vector_type(2))) float v2f;
typedef __attribute__((ext_vector_type(8))) float v8f;

#define TB 256
#define BM 32
#define BN 128
#define KC 32

#if defined(__has_builtin)
#  if __has_builtin(__builtin_amdgcn_global_load_async_to_lds_b128)
#    define HAS_ASYNC_LDS 1
#  endif
#endif
#ifndef HAS_ASYNC_LDS
#  define HAS_ASYNC_LDS 0
#endif

typedef int v4i __attribute__((vector_size(16)));
typedef __attribute__((address_space(1))) v4i as1_v4i;   // global
typedef __attribute__((address_space(3))) v4i as3_v4i;   // LDS

__device__ __forceinline__ void wait_async0() {
#if defined(__has_builtin) && __has_builtin(__builtin_amdgcn_s_wait_asynccnt)
    __builtin_amdgcn_s_wait_asynccnt(0);
#else
    asm volatile("s_wait_asynccnt 0x0" ::: "memory");
#endif
}

// C[m][n] = sum_k A[m][k]*B[k][n] + bias[n]
// Requires M%32==0, N%128==0, K%32==0, lda/ldb/ldc %4==0.
__global__ __launch_bounds__(TB) void wmma_gemm_f32(
    const float* __restrict__ A, const float* __restrict__ B,
    const float* __restrict__ bias, float* __restrict__ C,
    int M, int N, int K, int lda, int ldb, int ldc)
{
    __shared__ __align__(16) float As[2][BM][KC + 4];   // row stride 144B (16B mult)
    __shared__ __align__(16) float Bs[2][KC][BN + 4];   // row stride 528B (16B mult)

    const int tid  = threadIdx.x;
    const int lane = tid & 31;
    const int wave = tid >> 5;
    const int wm   = wave >> 2;   // 0..1  (16-row slab within BM)
    const int wn   = wave & 3;    // 0..3  (32-col slab within BN)

    const int rowBase = blockIdx.x * BM;
    const int colBase = blockIdx.y * BN;

    // per-thread staging coordinates (one float4 of A, four float4 of B)
    const int ar = tid >> 3;          // A row   (tid*4 / 32)
    const int ac = (tid & 7) * 4;     // A col

    v8f acc0 = {};
    v8f acc1 = {};

    const int mfrag = lane & 15;          // matrix row within 16
    const int koff  = (lane >> 4) * 2;    // lanes 16..31 hold K+2, K+3
    const int n0    = wn * 32;

    auto compute = [&](int buf) {
        #pragma unroll
        for (int ks = 0; ks < KC; ks += 4) {
            const int kk = ks + koff;
            v2f a, b0, b1;
            a.x  = As[buf][wm * 16 + mfrag][kk];
            a.y  = As[buf][wm * 16 + mfrag][kk + 1];
            b0.x = Bs[buf][kk][n0 + mfrag];
            b0.y = Bs[buf][kk + 1][n0 + mfrag];
            b1.x = Bs[buf][kk][n0 + 16 + mfrag];
            b1.y = Bs[buf][kk + 1][n0 + 16 + mfrag];
            acc0 = __builtin_amdgcn_wmma_f32_16x16x4_f32(
                false, a, false, b0, (short)0, acc0, false, false);
            acc1 = __builtin_amdgcn_wmma_f32_16x16x4_f32(
                false, a, false, b1, (short)0, acc1, false, false);
        }
    };

#if HAS_ASYNC_LDS
    // ---- CDNA5 async global->LDS DMA path (ASYNCcnt), double buffered ----
    auto stage_async = [&](int buf, int k0) {
        const float* gA = A + (size_t)(rowBase + ar) * lda + k0 + ac;
        __builtin_amdgcn_global_load_async_to_lds_b128(
            (as1_v4i*)gA, (as3_v4i*)&As[buf][ar][ac], 0, 0);
        #pragma unroll
        for (int i = 0; i < 4; ++i) {
            const int linear = tid * 4 + i * 1024;
            const int br = linear >> 7;
            const int bc = linear & 127;
            const float* gB = B + (size_t)(k0 + br) * ldb + colBase + bc;
            __builtin_amdgcn_global_load_async_to_lds_b128(
                (as1_v4i*)gB, (as3_v4i*)&Bs[buf][br][bc], 0, 0);
        }
    };

    stage_async(0, 0);
    wait_async0();
    __syncthreads();
    int cur = 0;
    for (int k0 = 0; k0 < K; k0 += KC) {
        const bool more = (k0 + KC) < K;
        if (more) stage_async(cur ^ 1, k0 + KC);   // DMA next chunk during WMMAs
        compute(cur);
        if (more) wait_async0();
        __syncthreads();
        cur ^= 1;
    }
#else
    // ---- fallback: register-staged double buffering ----
    float4 ra;
    float4 rb[4];
    auto gload = [&](int k0) {
        ra = *(const float4*)(A + (size_t)(rowBase + ar) * lda + k0 + ac);
        #pragma unroll
        for (int i = 0; i < 4; ++i) {
            const int linear = tid * 4 + i * 1024;
            const int br = linear >> 7;
            const int bc = linear & 127;
            rb[i] = *(const float4*)(B + (size_t)(k0 + br) * ldb + colBase + bc);
        }
    };
    auto lstore = [&](int buf) {
        *(float4*)&As[buf][ar][ac] = ra;
        #pragma unroll
        for (int i = 0; i < 4; ++i) {
            const int linear = tid * 4 + i * 1024;
            const int br = linear >> 7;
            const int bc = linear & 127;
            *(float4*)&Bs[buf][br][bc] = rb[i];
        }
    };

    gload(0);
    lstore(0);
    __syncthreads();
    int cur = 0;
    for (int k0 = 0; k0 < K; k0 += KC) {
        const bool more = (k0 + KC) < K;
        if (more) gload(k0 + KC);        // global loads in flight during WMMAs
        compute(cur);
        if (more) lstore(cur ^ 1);       // other buffer: safe while others read cur
        __syncthreads();
        cur ^= 1;
    }
#endif

    // ---- epilogue: C/D layout, VGPR r -> (M = r + 8*(lane>=16), N = lane&15) ----
    const int nlane = lane & 15;
    const int rAdd  = (lane >> 4) * 8;
    const int col0  = colBase + wn * 32 + nlane;
    const int col1  = col0 + 16;
    const float bv0 = bias[col0];
    const float bv1 = bias[col1];
    const int row0  = rowBase + wm * 16 + rAdd;
    #pragma unroll
    for (int r = 0; r < 8; ++r) {
        C[(size_t)(row0 + r) * ldc + col0] = acc0[r] + bv0;
        C[(size_t)(row0 + r) * ldc + col1] = acc1[r] + bv1;
    }
}

__device__ __forceinline__ float fast_sigmoid(float z) {
    return 1.0f / (1.0f + __expf(-z));
}
__device__ __forceinline__ float fast_tanh(float z) {
    return 2.0f / (1.0f + __expf(-2.0f * z)) - 1.0f;
}

#define DH_ 512
#define L_  4096

// One thread per (b, dh): fully independent diagonal scans.
__global__ __launch_bounds__(256) void lstm_scan(
    const float* __restrict__ ifgo,     // [B][L][DH*4], gate = fastest dim
    const float* __restrict__ kernel_h, // [4][DH]
    const float* __restrict__ bias_h,   // [4][DH]
    float* __restrict__ hs,             // [B][L][DH]
    float* __restrict__ c_out,          // [B][DH]
    float* __restrict__ h_out)          // [B][DH]
{
    const int tid = blockIdx.x * blockDim.x + threadIdx.x; // 0..4095
    const int b = tid >> 9;
    const int d = tid & (DH_ - 1);

    const float ki = kernel_h[0 * DH_ + d];
    const float kf = kernel_h[1 * DH_ + d];
    const float kg = kernel_h[2 * DH_ + d];
    const float ko = kernel_h[3 * DH_ + d];
    const float bi = bias_h[0 * DH_ + d];
    const float bf = bias_h[1 * DH_ + d];
    const float bg = bias_h[2 * DH_ + d];
    const float bo = bias_h[3 * DH_ + d];

    // each (b,l) row of ifgo is 512 float4's (gates contiguous per d)
    const float4* gp = (const float4*)(ifgo + (size_t)b * L_ * (4 * DH_)) + d;
    float* hp = hs + (size_t)b * L_ * DH_ + d;

    float c = 0.0f, h = 0.0f;
    float4 g = gp[0];
    for (int t = 0; t < L_; ++t) {
        float4 gn = g;
        if (t + 1 < L_) gn = gp[(size_t)(t + 1) * DH_];   // prefetch next step

        const float i_ = fast_sigmoid(ki * h + bi + g.x);
        const float f_ = fast_sigmoid(kf * h + bf + g.y);
        const float g_ = fast_tanh   (kg * h + bg + g.z);
        const float o_ = fast_sigmoid(ko * h + bo + g.w);
        c = f_ * c + i_ * g_;
        h = o_ * fast_tanh(c);

        hp[(size_t)t * DH_] = h;
        g = gn;
    }
    c_out[tid] = c;
    h_out[tid] = h;
}

extern "C" void kernel_launch(void* const* d_in, const int* in_sizes, int n_in,
                              void* d_out, int out_size, void* d_ws, size_t ws_size,
                              hipStream_t stream) {
    const float* x        = (const float*)d_in[0]; // (8, 4096, 256)
    const float* Wx       = (const float*)d_in[1]; // (256, 2048)
    const float* bx       = (const float*)d_in[2]; // (2048,)
    const float* kernel_h = (const float*)d_in[3]; // (4, 512)
    const float* bias_h   = (const float*)d_in[4]; // (4, 512)
    const float* Wo       = (const float*)d_in[5]; // (512, 256)
    const float* bo       = (const float*)d_in[6]; // (256,)

    const int M = 8 * 4096;          // 32768 rows (b*L + l)

    float* ifgo = (float*)d_ws;                       // 32768 x 2048 fp32
    float* hs   = ifgo + (size_t)M * 2048;            // 32768 x 512  fp32

    float* out   = (float*)d_out;                     // (8, 4096, 256)
    float* c_out = out + (size_t)M * 256;             // (8, 512) final c
    float* h_out = c_out + 8 * 512;                   // (8, 512) final h

    dim3 block(TB);

    // ifgo = x @ Wx + bx   : M=32768, N=2048, K=256
    dim3 grid1(M / BM, 2048 / BN);
    wmma_gemm_f32<<<grid1, block, 0, stream>>>(x, Wx, bx, ifgo,
                                               M, 2048, 256, 256, 2048, 2048);

    // sequential diagonal LSTM scan: 4096 independent (b,dh) threads
    lstm_scan<<<dim3(16), dim3(256), 0, stream>>>(ifgo, kernel_h, bias_h,
                                                  hs, c_out, h_out);

    // out = hs @ Wo + bo   : M=32768, N=256, K=512
    dim3 grid2(M / BM, 256 / BN);
    wmma_gemm_f32<<<grid2, block, 0, stream>>>(hs, Wo, bo, out,
                                               M, 256, 512, 512, 256, 256);
}